// DynamicRouting_38938173505610
// MI455X (gfx1250) — compile-verified
//
#include <hip/hip_runtime.h>
#include <hip/hip_bf16.h>

typedef __attribute__((ext_vector_type(2))) float v2f;
typedef __attribute__((ext_vector_type(8))) float v8f;

#define PLANE 65536           // 256*256
#define NBATCH 64
#define OUTX_ELEMS (64u * 4u * 65536u)   // out_x element count (out_y follows)

// ---------------------------------------------------------------------------
// Kernel 1: deterministic per-plane mean reduction.
// 512 blocks: blocks 0..255 reduce x planes (n*4+c), 256..511 reduce y planes.
// One block fully owns one 65536-element plane -> bitwise-deterministic order.
// ---------------------------------------------------------------------------
__global__ void dr_reduce_sums(const float* __restrict__ x,
                               const float* __restrict__ y,
                               float* __restrict__ ws) {
  int bid    = blockIdx.x;          // 0..511
  int tensor = bid >> 8;            // 0 = x, 1 = y
  int plane  = bid & 255;           // n*4 + c
  const float4* src =
      (const float4*)((tensor ? y : x) + (size_t)plane * PLANE);
  int t = threadIdx.x;              // 0..255
  float acc = 0.0f;
#pragma unroll 8
  for (int i = 0; i < 64; ++i) {    // 16384 float4 / 256 threads
    float4 v = src[t + i * 256];
    acc += (v.x + v.y) + (v.z + v.w);
  }
  __shared__ float red[256];
  red[t] = acc;
  __syncthreads();
  for (int s = 128; s > 0; s >>= 1) {
    if (t < s) red[t] += red[t + s];
    __syncthreads();
  }
  if (t == 0) ws[tensor * 256 + plane] = red[0];
}

// ---------------------------------------------------------------------------
// Kernel 2: gates + per-sample WMMA parameter blocks.
// ws layout (floats): [0,256) sum_x, [256,512) sum_y,
//                     [512 + n*144): Wx[16x4], Wy[16x4], bias[16]
// ---------------------------------------------------------------------------
__global__ void dr_make_params(const float* __restrict__ w_rf1, const float* __restrict__ b_rf1,
                               const float* __restrict__ w_rf2, const float* __restrict__ b_rf2,
                               const float* __restrict__ w_e1,  const float* __restrict__ b_e1,
                               const float* __restrict__ w_e2,  const float* __restrict__ b_e2,
                               const float* __restrict__ w_e3,  const float* __restrict__ b_e3,
                               const float* __restrict__ w_e4,  const float* __restrict__ b_e4,
                               float* __restrict__ ws) {
  int n = threadIdx.x;
  if (n >= NBATCH) return;
  const float inv = 1.0f / (float)PLANE;
  float p[4], q[4];
#pragma unroll
  for (int c = 0; c < 4; ++c) {
    p[c] = ws[n * 4 + c] * inv;          // mean of x[n,c]
    q[c] = ws[256 + n * 4 + c] * inv;    // mean of y[n,c]
  }
  float g1x0 = b_rf1[0], g1x1 = b_rf1[1];
  float g1y0 = b_rf1[0], g1y1 = b_rf1[1];
#pragma unroll
  for (int c = 0; c < 4; ++c) {
    g1x0 += w_rf1[c] * p[c];  g1x1 += w_rf1[4 + c] * p[c];
    g1y0 += w_rf1[c] * q[c];  g1y1 += w_rf1[4 + c] * q[c];
  }
  float gx0 = b_rf2[0] + w_rf2[0] * g1x0 + w_rf2[1] * g1x1;
  float gx1 = b_rf2[1] + w_rf2[2] * g1x0 + w_rf2[3] * g1x1;
  float gy0 = b_rf2[0] + w_rf2[0] * g1y0 + w_rf2[1] * g1y1;
  float gy1 = b_rf2[1] + w_rf2[2] * g1y0 + w_rf2[3] * g1y1;
  float mx0 = gx0 > 0.0f ? 1.0f : 0.0f, mx1 = gx1 > 0.0f ? 1.0f : 0.0f;
  float my0 = gy0 > 0.0f ? 1.0f : 0.0f, my1 = gy1 > 0.0f ? 1.0f : 0.0f;

  float* P = ws + 512 + n * 144;   // Wx[64] | Wy[64] | bias[16]
#pragma unroll
  for (int m = 0; m < 16; ++m) {
#pragma unroll
    for (int k = 0; k < 4; ++k) {
      float wxv = 0.0f, wyv = 0.0f;
      if (m < 4)       { wxv = mx0 * w_e1[m * 4 + k];       wyv = my0 * w_e3[m * 4 + k]; }
      else if (m < 12) { wxv = mx1 * w_e2[(m - 4) * 4 + k]; wyv = my1 * w_e4[(m - 4) * 4 + k]; }
      P[m * 4 + k]      = wxv;
      P[64 + m * 4 + k] = wyv;
    }
    float bv = 0.0f;
    if (m < 4)       bv = mx0 * b_e1[m]     + my0 * b_e3[m];
    else if (m < 12) bv = mx1 * b_e2[m - 4] + my1 * b_e4[m - 4];
    P[128 + m] = bv;
  }
}

// ---------------------------------------------------------------------------
// Kernel 3: masked 1x1-conv via chained v_wmma_f32_16x16x4_f32.
//   A = per-sample masked weights (M=16 out-ch, K=4 in-ch)
//   B = pixel tile (K=4 in-ch, N=16 pixels) -> coalesced b32 loads
//   D = 16 out-ch x 16 pixels; coalesced b32 stores, channels 12..15 dead.
// Grid: (64 pixel-blocks, 64 samples); 256 thr = 8 waves; 8 tiles/wave.
// ---------------------------------------------------------------------------
__global__ void __launch_bounds__(256)
dr_conv_wmma(const float* __restrict__ x, const float* __restrict__ y,
             const float* __restrict__ ws, float* __restrict__ out) {
  const int n    = blockIdx.y;
  const int lane = threadIdx.x & 31;
  const int wave = threadIdx.x >> 5;
  const int m    = lane & 15;     // A-row / B-column within lane-half
  const int hi   = lane >> 4;     // 0: K=0,1 & D rows 0-7 | 1: K=2,3 & D rows 8-15
  const int kh   = hi * 2;

  const float* P = ws + 512 + n * 144;
  v2f ax = { P[m * 4 + kh],      P[m * 4 + kh + 1] };
  v2f ay = { P[64 + m * 4 + kh], P[64 + m * 4 + kh + 1] };

  v8f cinit;
#pragma unroll
  for (int r = 0; r < 8; ++r) cinit[r] = P[128 + r + hi * 8];

  // Per-lane output channel bases (element offsets into concatenated d_out).
  int cb[8];
#pragma unroll
  for (int r = 0; r < 8; ++r) {
    int mm = r + hi * 8;
    int off = 0;
    if (mm < 4)       off = n * 4 * PLANE + mm * PLANE;                       // out_x ch mm
    else if (mm < 12) off = (int)OUTX_ELEMS + n * 8 * PLANE + (mm - 4) * PLANE; // out_y ch mm-4
    cb[r] = off;                                                              // mm>=12 unused
  }

  const float* xb = x + (size_t)n * 4 * PLANE + (size_t)kh * PLANE;
  const float* yb = y + (size_t)n * 4 * PLANE + (size_t)kh * PLANE;
  const int pbase = blockIdx.x * 1024 + wave * 128 + m;   // this lane's pixel column

#pragma unroll
  for (int t = 0; t < 8; ++t) {
    const int p = pbase + t * 16;
    v2f bx = { xb[p], xb[p + PLANE] };   // channels kh, kh+1 at pixel p
    v2f by = { yb[p], yb[p + PLANE] };

    v8f acc = __builtin_amdgcn_wmma_f32_16x16x4_f32(
        false, ax, false, bx, (short)0, cinit, false, false);
    acc = __builtin_amdgcn_wmma_f32_16x16x4_f32(
        false, ay, false, by, (short)0, acc, false, false);

#pragma unroll
    for (int r = 0; r < 4; ++r) out[cb[r] + p] = acc[r];   // ch 0-3 / 8-11
    if (hi == 0) {                                          // ch 4-7 (12-15 dead)
#pragma unroll
      for (int r = 4; r < 8; ++r) out[cb[r] + p] = acc[r];
    }
  }
}

extern "C" void kernel_launch(void* const* d_in, const int* in_sizes, int n_in,
                              void* d_out, int out_size, void* d_ws, size_t ws_size,
                              hipStream_t stream) {
  const float* x     = (const float*)d_in[0];
  const float* y     = (const float*)d_in[1];
  const float* w_rf1 = (const float*)d_in[2];
  const float* b_rf1 = (const float*)d_in[3];
  const float* w_rf2 = (const float*)d_in[4];
  const float* b_rf2 = (const float*)d_in[5];
  const float* w_e1  = (const float*)d_in[6];
  const float* b_e1  = (const float*)d_in[7];
  const float* w_e2  = (const float*)d_in[8];
  const float* b_e2  = (const float*)d_in[9];
  const float* w_e3  = (const float*)d_in[10];
  const float* b_e3  = (const float*)d_in[11];
  const float* w_e4  = (const float*)d_in[12];
  const float* b_e4  = (const float*)d_in[13];
  float* ws  = (float*)d_ws;
  float* out = (float*)d_out;

  dr_reduce_sums<<<512, 256, 0, stream>>>(x, y, ws);
  dr_make_params<<<1, 64, 0, stream>>>(w_rf1, b_rf1, w_rf2, b_rf2,
                                       w_e1, b_e1, w_e2, b_e2,
                                       w_e3, b_e3, w_e4, b_e4, ws);
  dr_conv_wmma<<<dim3(64, 64), 256, 0, stream>>>(x, y, ws, out);
}